// LinearStateSpace_27049704030517
// MI455X (gfx1250) — compile-verified
//
#include <hip/hip_runtime.h>

// ---------------------------------------------------------------------------
// Linear state-space scan as a two-level blocked parallel scan; all heavy
// FLOPs run through V_WMMA_F32_16X16X4_F32 (fp32 WMMA, gfx1250, wave32).
//
//   x_{t+1} = A x_t + B u_t ,  y_t = C x_t + D u_t ,  N = 32768 = 32*32*32
//
// This round: LDS-staged, double-buffered 64x64-macro-tile WMMA GEMM for all
// contiguous-operand GEMMs (the ~95 GFLOP of work); the simple direct-WMMA
// kernel is kept only for the strided z/x-pass steps (N=32, tiny).
// ---------------------------------------------------------------------------

typedef __attribute__((ext_vector_type(2))) float v2f;
typedef __attribute__((ext_vector_type(8))) float v8f;

#define ORDER 1024
#define USZ   128
#define YSZ   128
#define NSTEP 32768
#define K1    32              // level-1 time block length
#define NB    (NSTEP / K1)    // 1024 level-1 blocks
#define K2    32              // blocks per group
#define NG    (NB / K2)       // 32 groups
#define RB    (K1 * USZ)      // 4096 stacked rows per block

#define MT   64               // macro tile M
#define NT   64               // macro tile N
#define KC   16               // K chunk
#define KCP  18               // padded LDS leading dim (even -> b64 aligned)

// ---------------------------------------------------------------------------
// Fast GEMM (contiguous operands): C = A@B (+E).  M%64==0, N%64==0, K%16==0.
//   A[m*lda+k], B[k*ldb+n], E[m*lde+n], C[m*ldc+n]; batch via blockIdx.z.
// 256 threads = 8 waves; wave (wm,wn) owns two 16x16 tiles (n and n+16).
// Double-buffered LDS panels; global loads for chunk c+1 overlap WMMA on c.
// ---------------------------------------------------------------------------
__global__ __launch_bounds__(256)
void gemm_wmma_lds(const float* __restrict__ A, int lda, long sA,
                   const float* __restrict__ B, int ldb, long sB,
                   const float* __restrict__ E, int lde, long sE,
                   float*       __restrict__ C, int ldc, long sC,
                   int M, int N, int K)
{
  __shared__ float sAb[2][MT * KCP];   // sAb[buf][m*KCP + k]
  __shared__ float sBb[2][NT * KCP];   // sBb[buf][n*KCP + k]

  const int tilesN = N / NT;
  const int bm = (blockIdx.x / tilesN) * MT;
  const int bn = (blockIdx.x % tilesN) * NT;

  A += (long)blockIdx.z * sA;
  B += (long)blockIdx.z * sB;
  C += (long)blockIdx.z * sC;
  if (E) E += (long)blockIdx.z * sE;

  const int tidx = threadIdx.x;
  const int wave = tidx >> 5;
  const int lane = tidx & 31;
  const int wm   = (wave >> 1) * 16;          // 0,16,32,48
  const int wn   = (wave & 1) * 32;           // 0,32 ; tiles at wn and wn+16
  const int half = lane >> 4;                 // K-pair select
  const int l16  = lane & 15;

  // staging assignments (per chunk: A 64x16, B 16x64, 4 floats/thread each)
  const int ar  = tidx >> 2,  ak  = (tidx & 3) << 2;   // A: row, k-quad
  const int bk  = tidx >> 4,  bn4 = (tidx & 15) << 2;  // B: k-row, n-quad

  const int nc = K / KC;
  float4 ra, rb;

  // prologue: fetch chunk 0
  ra = *(const float4*)&A[(size_t)(bm + ar) * lda + ak];
  rb = *(const float4*)&B[(size_t)bk * ldb + bn + bn4];
  {
    float* pa = &sAb[0][ar * KCP + ak];
    pa[0] = ra.x; pa[1] = ra.y; pa[2] = ra.z; pa[3] = ra.w;
    float* pb = &sBb[0][bn4 * KCP + bk];
    pb[0] = rb.x; pb[KCP] = rb.y; pb[2 * KCP] = rb.z; pb[3 * KCP] = rb.w;
  }

  v8f acc0 = {0.f,0.f,0.f,0.f,0.f,0.f,0.f,0.f};
  v8f acc1 = {0.f,0.f,0.f,0.f,0.f,0.f,0.f,0.f};

  for (int c = 0; c < nc; ++c) {
    const int cur = c & 1;

    // issue global loads for chunk c+1 (overlaps WMMA below)
    if (c + 1 < nc) {
      ra = *(const float4*)&A[(size_t)(bm + ar) * lda + (c + 1) * KC + ak];
      rb = *(const float4*)&B[(size_t)((c + 1) * KC + bk) * ldb + bn + bn4];
    }
    // deep prefetch into L2 (global_prefetch_b8)
    if (c + 4 < nc) {
      __builtin_prefetch(&A[(size_t)(bm + ar) * lda + (c + 4) * KC + ak], 0, 1);
      __builtin_prefetch(&B[(size_t)((c + 4) * KC + bk) * ldb + bn + bn4], 0, 1);
    }

    __syncthreads();   // chunk c panels visible to all waves

#pragma unroll
    for (int kk = 0; kk < KC; kk += 4) {
      const int k0 = kk + half * 2;
      v2f a, b0, b1;
      a.x  = sAb[cur][(wm + l16) * KCP + k0];
      a.y  = sAb[cur][(wm + l16) * KCP + k0 + 1];
      b0.x = sBb[cur][(wn + l16) * KCP + k0];
      b0.y = sBb[cur][(wn + l16) * KCP + k0 + 1];
      b1.x = sBb[cur][(wn + 16 + l16) * KCP + k0];
      b1.y = sBb[cur][(wn + 16 + l16) * KCP + k0 + 1];
      acc0 = __builtin_amdgcn_wmma_f32_16x16x4_f32(false, a, false, b0,
                                                   (short)0, acc0, false, false);
      acc1 = __builtin_amdgcn_wmma_f32_16x16x4_f32(false, a, false, b1,
                                                   (short)0, acc1, false, false);
    }

    if (c + 1 < nc) {
      __syncthreads();   // all waves done reading buf (c+1)&1 two chunks ago
      const int nxt = (c + 1) & 1;
      float* pa = &sAb[nxt][ar * KCP + ak];
      pa[0] = ra.x; pa[1] = ra.y; pa[2] = ra.z; pa[3] = ra.w;
      float* pb = &sBb[nxt][bn4 * KCP + bk];
      pb[0] = rb.x; pb[KCP] = rb.y; pb[2 * KCP] = rb.z; pb[3 * KCP] = rb.w;
    }
  }

  // epilogue: C layout row = bm+wm+half*8+r, cols bn+wn+l16 and +16
  const int mC = bm + wm + half * 8;
  const int n0 = bn + wn + l16;
#pragma unroll
  for (int r = 0; r < 8; ++r) {
    float v0 = acc0[r], v1 = acc1[r];
    if (E) {
      v0 += E[(size_t)(mC + r) * lde + n0];
      v1 += E[(size_t)(mC + r) * lde + n0 + 16];
    }
    C[(size_t)(mC + r) * ldc + n0]      = v0;
    C[(size_t)(mC + r) * ldc + n0 + 16] = v1;
  }
}

// ---------------------------------------------------------------------------
// Direct GEMM with strided B/E/C columns (used for z-pass / x-pass, N=32).
// ---------------------------------------------------------------------------
__global__ void gemm_wmma_direct(const float* __restrict__ A, int lda,
                                 const float* __restrict__ B, int ldb, int csb,
                                 const float* __restrict__ E, int lde, int cse,
                                 float*       __restrict__ C, int ldc, int csc,
                                 int M, int N, int K)
{
  const int tilesN = N >> 4;
  const int tiles  = (M >> 4) * tilesN;
  const int tid    = blockIdx.x * 8 + (threadIdx.x >> 5);
  if (tid >= tiles) return;                 // wave-uniform
  const int tm = tid / tilesN;
  const int tn = tid % tilesN;

  const int lane = threadIdx.x & 31;
  const int half = lane >> 4;
  const int l16  = lane & 15;
  const size_t mA = (size_t)(tm * 16 + l16);
  const size_t nB = (size_t)(tn * 16 + l16);

  v8f acc = {0.f,0.f,0.f,0.f,0.f,0.f,0.f,0.f};
#pragma unroll 4
  for (int kk = 0; kk < K; kk += 4) {
    const int k0 = kk + half * 2;
    v2f a, b;
    a.x = A[mA * lda + k0];
    a.y = A[mA * lda + k0 + 1];
    b.x = B[(size_t)k0 * ldb + nB * csb];
    b.y = B[(size_t)(k0 + 1) * ldb + nB * csb];
    acc = __builtin_amdgcn_wmma_f32_16x16x4_f32(false, a, false, b,
                                                (short)0, acc, false, false);
  }
  const int mC = tm * 16 + half * 8;
#pragma unroll
  for (int r = 0; r < 8; ++r) {
    float v = acc[r];
    if (E) v += E[(size_t)(mC + r) * lde + nB * cse];
    C[(size_t)(mC + r) * ldc + nB * csc] = v;
  }
}

// --------------------------- small helper kernels --------------------------

__global__ void zero_f32(float* __restrict__ p, int n) {
  int i = blockIdx.x * 256 + threadIdx.x;
  if (i < n) p[i] = 0.f;
}

__global__ void copy_f32(float* __restrict__ d, const float* __restrict__ s, int n) {
  int i = blockIdx.x * 256 + threadIdx.x;
  if (i < n) d[i] = s[i];
}

// Place B (1024x128) into column-block 31 of W (1024x4096): W block j = A^{31-j} B
__global__ void scatter_B_W(float* __restrict__ W, const float* __restrict__ Bm) {
  int i = blockIdx.x * 256 + threadIdx.x;
  if (i >= ORDER * USZ) return;
  int row = i >> 7, k = i & 127;
  W[(size_t)row * RB + (K1 - 1) * USZ + k] = Bm[i];
}

// Ublk[(j*128+k)*NB + bb] = u_{bb*32+j}(k);  reference: U[k][t] = in0[k*NSTEP + t]
__global__ void build_Ublk(float* __restrict__ Ub, const float* __restrict__ U) {
  int i = blockIdx.x * 256 + threadIdx.x;
  if (i >= RB * NB) return;
  int bb = i % NB, rjk = i / NB;
  int j = rjk >> 7, k = rjk & 127;
  Ub[i] = U[(size_t)k * NSTEP + bb * K1 + j];
}

// Block-lower-triangular Toeplitz: T[j,i] block = (i==j) ? D : (i<j ? H_{j-1-i} : 0)
__global__ void build_T(float* __restrict__ T, const float* __restrict__ H,
                        const float* __restrict__ D) {
  int i = blockIdx.x * 256 + threadIdx.x;
  if (i >= RB * RB) return;
  int row = i / RB, col = i % RB;
  int j = row >> 7, r = row & 127, ib = col >> 7, k = col & 127;
  float v = 0.f;
  if (ib == j)      v = D[r * USZ + k];
  else if (ib < j)  v = H[((size_t)(j - 1 - ib) * YSZ + r) * USZ + k];
  T[i] = v;
}

// out[t*128 + r] = Y[(j*128+r)*NB + bb], t = bb*32 + j
__global__ void writeout(float* __restrict__ out, const float* __restrict__ Y) {
  int i = blockIdx.x * 256 + threadIdx.x;
  if (i >= NSTEP * YSZ) return;
  int t = i >> 7, r = i & 127;
  int bb = t >> 5, j = t & 31;
  out[i] = Y[(size_t)(j * YSZ + r) * NB + bb];
}

// y[i*ys] = (Mt @ x)[i] + d[i*ds]   (1024x1024 matvec, group-head chain)
__global__ void matvec_add(const float* __restrict__ Mt,
                           const float* __restrict__ x, int xs,
                           const float* __restrict__ d, int ds,
                           float* __restrict__ y, int ys) {
  int i = blockIdx.x * blockDim.x + threadIdx.x;
  if (i >= ORDER) return;
  const float* row = Mt + (size_t)i * ORDER;
  float s = 0.f;
  for (int k = 0; k < ORDER; ++k) s = fmaf(row[k], x[(size_t)k * xs], s);
  y[(size_t)i * ys] = s + d[(size_t)i * ds];
}

// ------------------------------- host side ---------------------------------

static void gemmF(hipStream_t s, int M, int N, int K,
                  const float* A, int lda, const float* B, int ldb,
                  float* C, int ldc,
                  const float* E = nullptr, int lde = 0,
                  int batch = 1, long sA = 0, long sB = 0, long sE = 0, long sC = 0)
{
  dim3 grid((M / MT) * (N / NT), 1, batch);
  gemm_wmma_lds<<<grid, dim3(256), 0, s>>>(A, lda, sA, B, ldb, sB,
                                           E, lde, sE, C, ldc, sC, M, N, K);
}

static void gemmS(hipStream_t s, int M, int N, int K,
                  const float* A, int lda,
                  const float* B, int ldb, int csb,
                  float* C, int ldc, int csc,
                  const float* E, int lde, int cse)
{
  int tiles = (M / 16) * (N / 16);
  dim3 grid((tiles + 7) / 8, 1, 1);
  gemm_wmma_direct<<<grid, dim3(256), 0, s>>>(A, lda, B, ldb, csb,
                                              E, lde, cse, C, ldc, csc, M, N, K);
}

static inline int gsz(int n) { return (n + 255) / 256; }

extern "C" void kernel_launch(void* const* d_in, const int* in_sizes, int n_in,
                              void* d_out, int out_size, void* d_ws, size_t ws_size,
                              hipStream_t stream)
{
  (void)in_sizes; (void)n_in; (void)out_size; (void)ws_size;
  const float* U  = (const float*)d_in[0];   // 128 x 32768 row-major (after reshape)
  const float* Am = (const float*)d_in[1];   // 1024 x 1024
  const float* Bm = (const float*)d_in[2];   // 1024 x 128
  const float* Cm = (const float*)d_in[3];   // 128  x 1024
  const float* Dm = (const float*)d_in[4];   // 128  x 128
  float* out = (float*)d_out;

  // ---- workspace carve (fp32), ~162 MB total ----
  float* p   = (float*)d_ws;
  float* CAb = p; p += (size_t)K1 * YSZ * ORDER;     // CAstack: [C; CA; ...; CA^31]
  float* Wb  = p; p += (size_t)ORDER * RB;           // [A^31 B, ..., A B, B]
  float* Tb  = p; p += (size_t)RB * RB;              // block-Toeplitz (4096x4096)
  float* Ub  = p; p += (size_t)RB * NB;              // stacked inputs per block
  float* Hb  = p; p += (size_t)(K1 - 1) * YSZ * USZ; // H_m = C A^m B, m=0..30
  float* Pb  = p; p += (size_t)ORDER * ORDER;        // A^32
  float* Qb  = p; p += (size_t)ORDER * ORDER;        // A^1024
  float* t1  = p; p += (size_t)ORDER * ORDER;
  float* t2  = p; p += (size_t)ORDER * ORDER;
  float* Cc  = p; p += (size_t)ORDER * NB;           // c_bb (1024 x 1024)
  float* z0  = p; p += (size_t)ORDER * NG;
  float* z1  = p; p += (size_t)ORDER * NG;
  float* X   = p; p += (size_t)ORDER * NB;           // block-start states
  float* Yb  = p; p += (size_t)RB * NB;              // stacked outputs

  // ---- init (X col 0 and z0 must be zero) ----
  zero_f32<<<gsz(ORDER * NB), 256, 0, stream>>>(X, ORDER * NB);
  zero_f32<<<gsz(ORDER * NG), 256, 0, stream>>>(z0, ORDER * NG);

  // ---- W = [A^31 B | ... | B] (chain inside W) ----
  scatter_B_W<<<gsz(ORDER * USZ), 256, 0, stream>>>(Wb, Bm);
  for (int j = 1; j < K1; ++j)   // block (31-j) = A * block (32-j)
    gemmF(stream, ORDER, USZ, ORDER, Am, ORDER,
          Wb + (size_t)(K1 - j) * USZ, RB,
          Wb + (size_t)(K1 - 1 - j) * USZ, RB);

  // ---- CAstack: CA[j] = CA[j-1] @ A ----
  copy_f32<<<gsz(YSZ * ORDER), 256, 0, stream>>>(CAb, Cm, YSZ * ORDER);
  for (int j = 1; j < K1; ++j)
    gemmF(stream, YSZ, ORDER, ORDER,
          CAb + (size_t)(j - 1) * YSZ * ORDER, ORDER, Am, ORDER,
          CAb + (size_t)j * YSZ * ORDER, ORDER);

  // ---- P = A^32, Q = A^1024 by repeated squaring ----
  gemmF(stream, ORDER, ORDER, ORDER, Am, ORDER, Am, ORDER, t1, ORDER); // A^2
  gemmF(stream, ORDER, ORDER, ORDER, t1, ORDER, t1, ORDER, t2, ORDER); // A^4
  gemmF(stream, ORDER, ORDER, ORDER, t2, ORDER, t2, ORDER, t1, ORDER); // A^8
  gemmF(stream, ORDER, ORDER, ORDER, t1, ORDER, t1, ORDER, t2, ORDER); // A^16
  gemmF(stream, ORDER, ORDER, ORDER, t2, ORDER, t2, ORDER, Pb, ORDER); // A^32
  gemmF(stream, ORDER, ORDER, ORDER, Pb, ORDER, Pb, ORDER, t1, ORDER); // A^64
  gemmF(stream, ORDER, ORDER, ORDER, t1, ORDER, t1, ORDER, t2, ORDER); // A^128
  gemmF(stream, ORDER, ORDER, ORDER, t2, ORDER, t2, ORDER, t1, ORDER); // A^256
  gemmF(stream, ORDER, ORDER, ORDER, t1, ORDER, t1, ORDER, t2, ORDER); // A^512
  gemmF(stream, ORDER, ORDER, ORDER, t2, ORDER, t2, ORDER, Qb, ORDER); // A^1024

  // ---- H_m = CA[m] @ B, m = 0..30 (batched over blockIdx.z) ----
  gemmF(stream, YSZ, USZ, ORDER, CAb, ORDER, Bm, USZ, Hb, USZ,
        nullptr, 0, K1 - 1, (long)YSZ * ORDER, 0, 0, (long)YSZ * USZ);

  build_T<<<gsz(RB * RB), 256, 0, stream>>>(Tb, Hb, Dm);
  build_Ublk<<<gsz(RB * NB), 256, 0, stream>>>(Ub, U);

  // ---- Phase 1: c_bb = W @ Ublk ----
  gemmF(stream, ORDER, NB, RB, Wb, RB, Ub, NB, Cc, NB);

  // ---- Phase 2a: z-pass  z <- P z + c_{32g+i}, batched over 32 groups ----
  for (int i = 0; i < K2; ++i) {
    const float* zin = (i & 1) ? z1 : z0;
    float*       zot = (i & 1) ? z0 : z1;
    gemmS(stream, ORDER, NG, ORDER, Pb, ORDER, zin, NG, 1, zot, NG, 1,
          Cc + i, NB, K2);
  }
  const float* dmat = z0;   // K2 even -> step 31 wrote z0: d_g per group

  // ---- Phase 2b: group-head chain  X[:,32(g+1)] = Q X[:,32g] + d_g ----
  for (int g = 0; g + 1 < NG; ++g)
    matvec_add<<<gsz(ORDER), 256, 0, stream>>>(Qb, X + (size_t)K2 * g, NB,
                                               dmat + g, NG,
                                               X + (size_t)K2 * (g + 1), NB);

  // ---- Phase 2c: x-pass  X[:,bb+1] = P X[:,bb] + c_bb, batched ----
  for (int i = 0; i + 1 < K2; ++i)
    gemmS(stream, ORDER, NG, ORDER, Pb, ORDER, X + i, NB, K2, X + (i + 1), NB, K2,
          Cc + i, NB, K2);

  // ---- Phase 3: Y = T @ Ublk ; Y += CAstack @ X ----
  gemmF(stream, RB, NB, RB, Tb, RB, Ub, NB, Yb, NB);
  gemmF(stream, RB, NB, ORDER, CAb, ORDER, X, NB, Yb, NB, Yb, NB);

  writeout<<<gsz(NSTEP * YSZ), 256, 0, stream>>>(out, Yb);
}